// EdgeGATPolicy_15144054686263
// MI455X (gfx1250) — compile-verified
//
#include <hip/hip_runtime.h>
#include <hip/hip_bf16.h>
#include <math.h>

// ---------------------------------------------------------------------------
// EdgeGAT policy, MI455X (gfx1250, wave32).
// Key transform: edge_h = edge_tokens @ W_edge.T is never materialized.
//   att_raw_pre = edge_tokens @ u + c[g],  u = W_edge^T att_vec, c = q @ (W_query^T att_vec)
//   pooled      = (segment_sum(edge_tokens)) @ W_edge^T / count
// -> single 512MB streaming pass (HBM-bound, ~22us floor) + tiny WMMA GEMMs.
// ---------------------------------------------------------------------------

typedef float v2f __attribute__((ext_vector_type(2)));
typedef float v8f __attribute__((ext_vector_type(8)));
typedef float f4  __attribute__((ext_vector_type(4)));

#define HDIM 256

#if __has_builtin(__builtin_amdgcn_wmma_f32_16x16x4_f32)
#define WMMA_F32_4(a, b, c) \
  __builtin_amdgcn_wmma_f32_16x16x4_f32(false, (a), false, (b), (short)0, (c), false, false)
#else
static __device__ inline v8f wmma_fallback(v2f a, v2f b, v8f c) {
  for (int i = 0; i < 8; ++i) c[i] += a.x * b.x + a.y * b.y;
  return c;
}
#define WMMA_F32_4(a, b, c) wmma_fallback((a), (b), (c))
#endif

// Monotonic float<->uint encoding so atomicMax(uint) implements float max.
static __device__ inline unsigned fenc(float f) {
  unsigned u = __float_as_uint(f);
  return (u & 0x80000000u) ? ~u : (u | 0x80000000u);
}
static __device__ inline float fdec(unsigned u) {
  return (u & 0x80000000u) ? __uint_as_float(u & 0x7fffffffu) : __uint_as_float(~u);
}

// ---------------------------------------------------------------------------
// u[j] = sum_i W_edge[i][j] * att_vec[i]   (block 0)
// v[j] = sum_i W_query[i][j] * att_vec[i]  (block 1)
__global__ void k_uv(const float* __restrict__ We, const float* __restrict__ Wq,
                     const float* __restrict__ av, float* __restrict__ u,
                     float* __restrict__ v) {
  const float* W = (blockIdx.x == 0) ? We : Wq;
  float* out = (blockIdx.x == 0) ? u : v;
  int j = threadIdx.x;
  float acc = 0.f;
  for (int i = 0; i < HDIM; ++i) acc = fmaf(W[i * HDIM + j], av[i], acc);
  out[j] = acc;
}

// c[g] = dot(question_tokens[g], v)
__global__ void k_c(const float* __restrict__ q, const float* __restrict__ v,
                    float* __restrict__ c, int G) {
  int g = threadIdx.x;
  if (g >= G) return;
  float acc = 0.f;
  for (int k = 0; k < HDIM; ++k) acc = fmaf(q[(size_t)g * HDIM + k], v[k], acc);
  c[g] = acc;
}

// zero S[G*H], segsum[G], cnt[G], segmax_u[G]
__global__ void k_init(float* __restrict__ S, float* __restrict__ segsum,
                       unsigned* __restrict__ segmax, int* __restrict__ cnt,
                       int G) {
  int t = blockIdx.x * blockDim.x + threadIdx.x;
  int n = G * HDIM;
  if (t < n) S[t] = 0.f;
  if (t < G) { segsum[t] = 0.f; segmax[t] = 0u; cnt[t] = 0; }
}

// ---------------------------------------------------------------------------
// Main streaming pass. Block = 256 threads (8 waves), 256 edges per block,
// wave-per-edge.  Accumulates per-block S / max / count in LDS (edge_batch is
// sorted, so a block spans few graphs), then flushes with global atomics.
#define MAXG 8
__global__ void __launch_bounds__(256)
k_main(const float* __restrict__ tok, const int* __restrict__ eb,
       const unsigned char* __restrict__ sel, const float* __restrict__ u,
       const float* __restrict__ c, float* __restrict__ att_raw,
       float* __restrict__ S, unsigned* __restrict__ segmax,
       int* __restrict__ cnt, int E) {
  __shared__ float u_s[HDIM];
  __shared__ float acc[MAXG][HDIM];
  __shared__ unsigned max_s[MAXG];
  __shared__ int cnt_s[MAXG];
  __shared__ int s_glo, s_range, s_wide;

  const int t = threadIdx.x;
  const int wv = t >> 5, lane = t & 31;
  const int e0 = blockIdx.x * 256;

  u_s[t] = u[t];
  for (int i = t; i < MAXG * HDIM; i += 256) (&acc[0][0])[i] = 0.f;
  if (t < MAXG) { max_s[t] = 0u; cnt_s[t] = 0; }
  if (t == 0) {
    int elo = min(e0, E - 1);
    int ehi = min(e0 + 255, E - 1);
    int glo = eb[elo], ghi = eb[ehi];
    s_glo = glo;
    s_range = ghi - glo + 1;
    s_wide = (ghi - glo + 1 > MAXG) ? 1 : 0;
  }
  __syncthreads();

  const int glo = s_glo, wide = s_wide;
  const int col = lane << 2;

  for (int i = 0; i < 32; ++i) {
    int e = e0 + wv * 32 + i;
    if (e >= E) break;
    const float* row = tok + (size_t)e * HDIM;
    if (i + 2 < 32) __builtin_prefetch(row + 2 * HDIM + col, 0, 1);

    f4 f0 = __builtin_nontemporal_load((const f4*)(row + col));
    f4 f1 = __builtin_nontemporal_load((const f4*)(row + 128 + col));
    f4 u0 = *(const f4*)(u_s + col);
    f4 u1 = *(const f4*)(u_s + 128 + col);
    float d = f0.x * u0.x + f0.y * u0.y + f0.z * u0.z + f0.w * u0.w +
              f1.x * u1.x + f1.y * u1.y + f1.z * u1.z + f1.w * u1.w;
    for (int off = 16; off; off >>= 1) d += __shfl_xor(d, off, 32);

    int g = eb[e];
    int cand = sel[e] ? 0 : 1;
    float att = d + c[g];
    att = (att > 0.f) ? att : 0.2f * att;       // LeakyReLU(0.2)
    att += 0.5f * (float)cand;                  // frontier bonus
    if (lane == 0) att_raw[e] = att;

    if (!wide) {
      int gi = g - glo;
      atomicAdd(&acc[gi][col + 0], f0.x);
      atomicAdd(&acc[gi][col + 1], f0.y);
      atomicAdd(&acc[gi][col + 2], f0.z);
      atomicAdd(&acc[gi][col + 3], f0.w);
      atomicAdd(&acc[gi][128 + col + 0], f1.x);
      atomicAdd(&acc[gi][128 + col + 1], f1.y);
      atomicAdd(&acc[gi][128 + col + 2], f1.z);
      atomicAdd(&acc[gi][128 + col + 3], f1.w);
      if (lane == 0) { atomicMax(&max_s[gi], fenc(att)); atomicAdd(&cnt_s[gi], 1); }
    } else {
      size_t base = (size_t)g * HDIM;
      atomicAdd(&S[base + col + 0], f0.x);
      atomicAdd(&S[base + col + 1], f0.y);
      atomicAdd(&S[base + col + 2], f0.z);
      atomicAdd(&S[base + col + 3], f0.w);
      atomicAdd(&S[base + 128 + col + 0], f1.x);
      atomicAdd(&S[base + 128 + col + 1], f1.y);
      atomicAdd(&S[base + 128 + col + 2], f1.z);
      atomicAdd(&S[base + 128 + col + 3], f1.w);
      if (lane == 0) { atomicMax(&segmax[g], fenc(att)); atomicAdd(&cnt[g], 1); }
    }
  }
  __syncthreads();
  if (!wide) {
    for (int gi = 0; gi < s_range; ++gi) {
      float val = acc[gi][t];
      if (val != 0.f) atomicAdd(&S[(size_t)(glo + gi) * HDIM + t], val);
      if (t == 0 && cnt_s[gi] > 0) {
        atomicMax(&segmax[glo + gi], max_s[gi]);
        atomicAdd(&cnt[glo + gi], cnt_s[gi]);
      }
    }
  }
}

// pass 2: segsum[g] += cand * exp(att - max[g])
__global__ void k_exps(const float* __restrict__ att_raw, const int* __restrict__ eb,
                       const unsigned char* __restrict__ sel,
                       const unsigned* __restrict__ segmax,
                       float* __restrict__ segsum, int E) {
  int e = blockIdx.x * blockDim.x + threadIdx.x;
  if (e >= E) return;
  if (sel[e]) return;                       // not a candidate -> contributes 0
  int g = eb[e];
  float x = expf(att_raw[e] - fdec(segmax[g]));
  atomicAdd(&segsum[g], x);
}

// pass 3: edge_logits = log(max(prob, eps))
__global__ void k_logits(const float* __restrict__ att_raw, const int* __restrict__ eb,
                         const unsigned char* __restrict__ sel,
                         const unsigned* __restrict__ segmax,
                         const float* __restrict__ segsum,
                         float* __restrict__ out, int E) {
  const float eps = 1.1920929e-7f;          // float32 machine eps
  int e = blockIdx.x * blockDim.x + threadIdx.x;
  if (e >= E) return;
  int g = eb[e];
  float p = 0.f;
  if (!sel[e]) p = expf(att_raw[e] - fdec(segmax[g])) / fmaxf(segsum[g], eps);
  out[e] = logf(fmaxf(p, eps));
}

// ---------------------------------------------------------------------------
// C[M,N] = A[M,K] * B[N,K]^T using v_wmma_f32_16x16x4_f32, wave per 16x16 tile.
#define GEMM_WAVES 4
__global__ void __launch_bounds__(GEMM_WAVES * 32)
k_gemm16(const float* __restrict__ A, const float* __restrict__ B,
         float* __restrict__ C, int M, int N, int K, int lda, int ldb, int ldc) {
  int wave = threadIdx.x >> 5;
  int lane = threadIdx.x & 31;
  int tiles_n = N >> 4;
  int total = (M >> 4) * tiles_n;
  int tile = blockIdx.x * GEMM_WAVES + wave;
  if (tile >= total) return;                // whole wave exits together
  int tm = tile / tiles_n, tn = tile % tiles_n;
  int r = lane & 15, s = lane >> 4;         // s selects K-pair (0: k0..1, 1: k2..3)

  const float* arow = A + (size_t)(tm * 16 + r) * lda;
  const float* brow = B + (size_t)(tn * 16 + r) * ldb;
  v8f acc = {0.f, 0.f, 0.f, 0.f, 0.f, 0.f, 0.f, 0.f};
  for (int k = 0; k < K; k += 4) {
    v2f a, b;
    a.x = arow[k + 2 * s]; a.y = arow[k + 2 * s + 1];
    b.x = brow[k + 2 * s]; b.y = brow[k + 2 * s + 1];
    acc = WMMA_F32_4(a, b, acc);
  }
  // C/D layout: lanes 0-15 -> M=v, lanes 16-31 -> M=8+v; N = lane%16
  float* crow = C + (size_t)(tm * 16 + s * 8) * ldc + tn * 16 + r;
  for (int v = 0; v < 8; ++v) crow[(size_t)v * ldc] = acc[v];
}

// ---------------------------------------------------------------------------
// per graph: pooled = pooled_raw/denom (also emitted), LayerNorm over [pooled|q]
__global__ void __launch_bounds__(256)
k_stop_ln(const float* __restrict__ pooled_raw, const int* __restrict__ cnt,
          const float* __restrict__ q, const float* __restrict__ ln_g,
          const float* __restrict__ ln_b, float* __restrict__ pooled_out,
          float* __restrict__ xn) {
  __shared__ float red[256];
  int g = blockIdx.x, t = threadIdx.x;
  float denom = (float)max(cnt[g], 1);
  float x0 = pooled_raw[(size_t)g * HDIM + t] / denom;
  float x1 = q[(size_t)g * HDIM + t];
  pooled_out[(size_t)g * HDIM + t] = x0;

  red[t] = x0 + x1;
  __syncthreads();
  for (int off = 128; off; off >>= 1) {
    if (t < off) red[t] += red[t + off];
    __syncthreads();
  }
  float mu = red[0] / (2.0f * HDIM);
  __syncthreads();
  red[t] = (x0 - mu) * (x0 - mu) + (x1 - mu) * (x1 - mu);
  __syncthreads();
  for (int off = 128; off; off >>= 1) {
    if (t < off) red[t] += red[t + off];
    __syncthreads();
  }
  float rstd = rsqrtf(red[0] / (2.0f * HDIM) + 1e-5f);
  xn[(size_t)g * 2 * HDIM + t]        = (x0 - mu) * rstd * ln_g[t] + ln_b[t];
  xn[(size_t)g * 2 * HDIM + HDIM + t] = (x1 - mu) * rstd * ln_g[HDIM + t] + ln_b[HDIM + t];
}

// per graph: stop_logit = dot(GELU(h1pre + b1), W2) + b2
__global__ void __launch_bounds__(256)
k_stop_fin(const float* __restrict__ h1pre, const float* __restrict__ b1,
           const float* __restrict__ W2, const float* __restrict__ b2,
           float* __restrict__ out) {
  __shared__ float red[256];
  int g = blockIdx.x, t = threadIdx.x;
  float h = h1pre[(size_t)g * HDIM + t] + b1[t];
  float gelu = 0.5f * h * (1.0f + erff(h * 0.70710678118654752f));  // exact GELU
  red[t] = gelu * W2[t];
  __syncthreads();
  for (int off = 128; off; off >>= 1) {
    if (t < off) red[t] += red[t + off];
    __syncthreads();
  }
  if (t == 0) out[g] = red[0] + b2[0];
}

// ---------------------------------------------------------------------------
extern "C" void kernel_launch(void* const* d_in, const int* in_sizes, int n_in,
                              void* d_out, int out_size, void* d_ws, size_t ws_size,
                              hipStream_t stream) {
  const float* edge_tokens = (const float*)d_in[0];
  const float* question    = (const float*)d_in[1];
  const int*   edge_batch  = (const int*)d_in[2];
  const unsigned char* sel = (const unsigned char*)d_in[3];
  const float* W_edge  = (const float*)d_in[4];
  const float* W_query = (const float*)d_in[5];
  const float* att_vec = (const float*)d_in[6];
  const float* ln_g = (const float*)d_in[7];
  const float* ln_b = (const float*)d_in[8];
  const float* W1 = (const float*)d_in[9];
  const float* b1 = (const float*)d_in[10];
  const float* W2 = (const float*)d_in[11];
  const float* b2 = (const float*)d_in[12];

  const int E = in_sizes[2];
  const int H = HDIM;
  const int G = in_sizes[1] / H;

  // workspace layout (floats)
  float* wsf = (float*)d_ws;
  float*    u       = wsf;                           // H
  float*    v       = u + H;                         // H
  float*    c       = v + H;                         // G
  float*    segsum  = c + 256;                       // G
  unsigned* segmax  = (unsigned*)(segsum + 256);     // G
  int*      cnt     = (int*)(segmax + 256);          // G
  float*    S       = (float*)(cnt + 256);           // G*H
  float*    pooled_raw = S + (size_t)G * H;          // G*H
  float*    xn      = pooled_raw + (size_t)G * H;    // G*2H
  float*    h1      = xn + (size_t)G * 2 * H;        // G*H
  float*    att_raw = h1 + (size_t)G * H;            // E

  float* out_edge   = (float*)d_out;                 // [E]
  float* out_stop   = out_edge + E;                  // [G]
  float* out_pooled = out_stop + G;                  // [G*H]

  // 1) tiny precomputes
  k_uv<<<2, 256, 0, stream>>>(W_edge, W_query, att_vec, u, v);
  k_c<<<1, 256, 0, stream>>>(question, v, c, G);
  k_init<<<(G * H + 255) / 256, 256, 0, stream>>>(S, segsum, segmax, cnt, G);

  // 2) single 512MB streaming pass over edge_tokens
  int mblocks = (E + 255) / 256;
  k_main<<<mblocks, 256, 0, stream>>>(edge_tokens, edge_batch, sel, u, c,
                                      att_raw, S, segmax, cnt, E);

  // 3) segment softmax over att_raw (2MB)
  k_exps<<<mblocks, 256, 0, stream>>>(att_raw, edge_batch, sel, segmax, segsum, E);
  k_logits<<<mblocks, 256, 0, stream>>>(att_raw, edge_batch, sel, segmax, segsum,
                                        out_edge, E);

  // 4) pooled = (S @ W_edge^T) / denom   (WMMA)
  int tiles = (G / 16) * (H / 16);
  k_gemm16<<<(tiles + GEMM_WAVES - 1) / GEMM_WAVES, GEMM_WAVES * 32, 0, stream>>>(
      S, W_edge, pooled_raw, G, H, H, H, H, H);
  k_stop_ln<<<G, 256, 0, stream>>>(pooled_raw, cnt, question, ln_g, ln_b,
                                   out_pooled, xn);

  // 5) stop head GEMM: h1pre = xn @ W1^T, K = 2H   (WMMA)
  k_gemm16<<<(tiles + GEMM_WAVES - 1) / GEMM_WAVES, GEMM_WAVES * 32, 0, stream>>>(
      xn, W1, h1, G, H, 2 * H, 2 * H, 2 * H, H);
  k_stop_fin<<<G, 256, 0, stream>>>(h1, b1, W2, b2, out_stop);
}